// Spiralconv_78503412236712
// MI455X (gfx1250) — compile-verified
//
#include <hip/hip_runtime.h>
#include <hip/hip_bf16.h>
#include <stdint.h>

typedef __attribute__((ext_vector_type(2))) float v2f;
typedef __attribute__((ext_vector_type(8))) float v8f;
typedef __attribute__((ext_vector_type(4))) unsigned int u32x4;
typedef __attribute__((ext_vector_type(8))) unsigned int u32x8;

#define BATCH   4
#define NNODES  50000
#define CIN     128
#define SEQ     9
#define COUT    128
#define K_TOT   (CIN * SEQ)     /* 1152 */
#define PADK    (K_TOT + 4)     /* 1156 dwords: stride % 64 == 4 -> conflict-free ds reads */
#define MTILE   32              /* nodes per tile (2 WMMA M-tiles) */
#define NT      ((NNODES + MTILE - 1) / MTILE)   /* 1563 */
#define PERSIST 256             /* persistent workgroups per batch */

// ---- TDM gather-mode descriptor: 9 rows (node neighbors) x 128 f32 -> LDS ----
// Group0: count=1, gather_mode=1, 16-bit indices, lds_addr, global_addr, type=2
// Group1: data_size=4B, tensor=[dim0=128, dim1=50000], tile_dim0=128,
//         tile_dim1=9 (valid indices), tensor_dim0_stride=128
// Group2/3: 9 packed 16-bit row indices
__device__ __forceinline__ void tdm_issue_node(uint32_t ldsByte, uint64_t gaddr,
                                               const int* __restrict__ ip)
{
    uint32_t i0 = (uint32_t)__builtin_amdgcn_readfirstlane(ip[0]);
    uint32_t i1 = (uint32_t)__builtin_amdgcn_readfirstlane(ip[1]);
    uint32_t i2 = (uint32_t)__builtin_amdgcn_readfirstlane(ip[2]);
    uint32_t i3 = (uint32_t)__builtin_amdgcn_readfirstlane(ip[3]);
    uint32_t i4 = (uint32_t)__builtin_amdgcn_readfirstlane(ip[4]);
    uint32_t i5 = (uint32_t)__builtin_amdgcn_readfirstlane(ip[5]);
    uint32_t i6 = (uint32_t)__builtin_amdgcn_readfirstlane(ip[6]);
    uint32_t i7 = (uint32_t)__builtin_amdgcn_readfirstlane(ip[7]);
    uint32_t i8 = (uint32_t)__builtin_amdgcn_readfirstlane(ip[8]);

    u32x4 g0;
    g0[0] = 0x80000001u;                                  // gather_mode=1, idx16, count=1
    g0[1] = ldsByte;                                      // lds_addr
    g0[2] = (uint32_t)gaddr;                              // global_addr[31:0]
    g0[3] = (uint32_t)((gaddr >> 32) & 0x01FFFFFFu) | (2u << 30);  // addr[56:32] | type=2

    u32x8 g1;
    g1[0] = 0x00020000u;                                  // data_size = 2 (4 bytes)
    g1[1] = ((uint32_t)CIN) << 16;                        // tensor_dim0 = 128 (lo16 @bit48)
    g1[2] = (((uint32_t)NNODES) & 0xFFFFu) << 16;         // tensor_dim1 lo16 @bit80
    g1[3] = ((((uint32_t)NNODES) >> 16) & 0xFFFFu)        // tensor_dim1 hi16
          | (((uint32_t)CIN) << 16);                      // tile_dim0 = 128
    g1[4] = (uint32_t)SEQ;                                // tile_dim1 = 9 valid indices
    g1[5] = (uint32_t)CIN;                                // tensor_dim0_stride = 128
    g1[6] = 0u;
    g1[7] = 0u;

    u32x4 g2, g3;
    g2[0] = (i0 & 0xFFFFu) | (i1 << 16);
    g2[1] = (i2 & 0xFFFFu) | (i3 << 16);
    g2[2] = (i4 & 0xFFFFu) | (i5 << 16);
    g2[3] = (i6 & 0xFFFFu) | (i7 << 16);
    g3[0] = (i8 & 0xFFFFu);
    g3[1] = 0u; g3[2] = 0u; g3[3] = 0u;

    asm volatile("tensor_load_to_lds %0, %1, %2, %3"
                 :: "s"(g0), "s"(g1), "s"(g2), "s"(g3)
                 : "memory");
}

// grid = (PERSIST, BATCH), block = 256 (8 waves).  Persistent WGs walk node
// tiles with double-buffered LDS; TDM gathers tile i+1 while WMMA computes tile i.
__global__ __launch_bounds__(256)
void spiralconv_tdm_wmma(const float* __restrict__ x,
                         const int*   __restrict__ indices,
                         const float* __restrict__ W,
                         const float* __restrict__ bias,
                         float*       __restrict__ out)
{
    extern __shared__ float ldsA[];   // 2 * MTILE * PADK floats = 295,936 B

    const int b    = blockIdx.y;
    const int tid  = threadIdx.x;
    const int wave = __builtin_amdgcn_readfirstlane(tid >> 5);  // 0..7
    const int lane = tid & 31;

    const float*   xb      = x + (size_t)b * NNODES * CIN;
    const uint64_t gaddr   = (uint64_t)(uintptr_t)xb;
    const uint32_t ldsBase = (uint32_t)(uintptr_t)&ldsA[0];

    // Wave issues TDM gathers for tile rows m = 4*wave .. 4*wave+3.
    auto issue_tile = [&](int t, int buf) {
        const int node0 = t * MTILE;
        #pragma unroll
        for (int j = 0; j < 4; ++j) {
            const int m = wave * 4 + j;
            int nm = node0 + m;
            if (nm > NNODES - 1) nm = NNODES - 1;       // tail clamp
            const int* ip = indices + (size_t)nm * SEQ;
            const uint32_t ldsByte = ldsBase + (uint32_t)((buf * MTILE + m) * PADK) * 4u;
            tdm_issue_node(ldsByte, gaddr, ip);
        }
    };

    // Per-wave WMMA tile setup: column tile n0 = 16*wave.
    const int lhalf = lane >> 4;       // 0: K=k0,k0+1  1: K=k0+2,k0+3
    const int lrow  = lane & 15;       // M (A) / N (B)
    const int n0    = wave * 16;
    const float* bBase = W + (size_t)(n0 + lrow) * K_TOT + 2 * lhalf;
    const float  bl    = bias[n0 + lrow];

    int t   = blockIdx.x;
    int buf = 0;
    if (t < NT) issue_tile(t, 0);      // prologue

    for (; t < NT; t += PERSIST) {
        __builtin_amdgcn_s_wait_tensorcnt(0);   // this wave's TDM ops for tile t done
        __syncthreads();                        // all waves' rows visible in buf

        const int tn = t + PERSIST;
        if (tn < NT) issue_tile(tn, buf ^ 1);   // overlap: gather next tile via TDM

        // ---- compute: two 16x16 tiles, 4 independent WMMA chains ----
        const float* aBase0 = &ldsA[(size_t)buf * MTILE * PADK + lrow * PADK + 2 * lhalf];
        const float* aBase1 = aBase0 + 16 * PADK;

        v8f accA0 = {bl, bl, bl, bl, bl, bl, bl, bl};   // tile0, even K-steps (+bias)
        v8f accA1 = {0, 0, 0, 0, 0, 0, 0, 0};           // tile0, odd  K-steps
        v8f accB0 = accA0;                              // tile1, even
        v8f accB1 = accA1;                              // tile1, odd

        #pragma unroll 4
        for (int k0 = 0; k0 < K_TOT; k0 += 8) {
            const v2f a00 = *reinterpret_cast<const v2f*>(aBase0 + k0);
            const v2f a01 = *reinterpret_cast<const v2f*>(aBase0 + k0 + 4);
            const v2f a10 = *reinterpret_cast<const v2f*>(aBase1 + k0);
            const v2f a11 = *reinterpret_cast<const v2f*>(aBase1 + k0 + 4);
            const v2f b0  = *reinterpret_cast<const v2f*>(bBase  + k0);
            const v2f b1  = *reinterpret_cast<const v2f*>(bBase  + k0 + 4);
            accA0 = __builtin_amdgcn_wmma_f32_16x16x4_f32(false, a00, false, b0, (short)0, accA0, false, false);
            accB0 = __builtin_amdgcn_wmma_f32_16x16x4_f32(false, a10, false, b0, (short)0, accB0, false, false);
            accA1 = __builtin_amdgcn_wmma_f32_16x16x4_f32(false, a01, false, b1, (short)0, accA1, false, false);
            accB1 = __builtin_amdgcn_wmma_f32_16x16x4_f32(false, a11, false, b1, (short)0, accB1, false, false);
        }
        const v8f acc0 = accA0 + accA1;
        const v8f acc1 = accB0 + accB1;

        // ---- store D tiles (C/D layout: VGPR r -> M=r+8*lhalf; N=lrow) ----
        const int node0 = t * MTILE;
        float* ob = out + ((size_t)b * NNODES + node0) * COUT + n0 + lrow;
        #pragma unroll
        for (int r = 0; r < 8; ++r) {
            const int row0 = r + 8 * lhalf;
            if (node0 + row0 < NNODES) ob[(size_t)row0 * COUT] = acc0[r];
            const int row1 = row0 + 16;
            if (node0 + row1 < NNODES) ob[(size_t)row1 * COUT] = acc1[r];
        }

        buf ^= 1;
        // Safety of refill: tile t+2*PERSIST is issued only after the next
        // iteration's barrier, i.e. after every wave finished computing tile t.
    }
}

extern "C" void kernel_launch(void* const* d_in, const int* in_sizes, int n_in,
                              void* d_out, int out_size, void* d_ws, size_t ws_size,
                              hipStream_t stream) {
    const float* x    = (const float*)d_in[0];
    const int*   idx  = (const int*)  d_in[1];
    const float* W    = (const float*)d_in[2];
    const float* bias = (const float*)d_in[3];
    float*       out  = (float*)d_out;

    dim3 grid(PERSIST, BATCH);                                   // 256 x 4 persistent WGs
    const size_t ldsBytes = 2u * MTILE * PADK * sizeof(float);   // 295,936 B
    spiralconv_tdm_wmma<<<grid, dim3(256), ldsBytes, stream>>>(x, idx, W, bias, out);
}